// Model_4217657884945
// MI455X (gfx1250) — compile-verified
//
#include <hip/hip_runtime.h>
#include <math.h>

typedef __attribute__((ext_vector_type(2))) float v2f;
typedef __attribute__((ext_vector_type(8))) float v8f;

#define BETA_ 0.3f
#define INV_SQRT2_ 0.70710678118654752440f

// ---- workspace layout (float offsets) ----
static constexpr size_t O_XW   = 0;                       // 128*1024*128
static constexpr size_t O_WCT  = O_XW   + 16777216ull;    // 3*1024*1024
static constexpr size_t O_H2   = O_WCT  + 3145728ull;     // 128*1024*16
static constexpr size_t O_CV   = O_H2   + 2097152ull;     // 128*1024*16
static constexpr size_t O_MT   = O_CV   + 2097152ull;     // 128*336*16
static constexpr size_t O_MEAN = O_MT   + 688128ull;      // 128*128
static constexpr size_t O_STD  = O_MEAN + 16384ull;       // 128*128
static constexpr size_t O_RH   = O_STD  + 16384ull;       // 336
static constexpr size_t O_RT   = O_RH   + 336ull;         // 336
static constexpr size_t O_RBC  = O_RT   + 336ull;         // 336
// total ~99.4 MB

static __device__ inline v8f wmma4(v2f a, v2f b, v8f c) {
    // D(16x16,f32) += A(16x4,f32) * B(4x16,f32)
    return __builtin_amdgcn_wmma_f32_16x16x4_f32(
        false, a, false, b, (short)0, c, false, false);
}

// ---------- 1. Haar level-1 : x(B,1024,128) -> xw(B,1024,128) ----------
__global__ __launch_bounds__(256) void k_haar(const float* __restrict__ x,
                                              float* __restrict__ xw) {
    size_t idx = (size_t)blockIdx.x * 256 + threadIdx.x; // B*512*C threads
    size_t c  = idx & 127;
    size_t sh = (idx >> 7) & 511;
    size_t b  = idx >> 16;
    const float* xb = x + b * 131072 + sh * 256 + c;
    float xe = xb[0], xo = xb[128];
    float* o = xw + b * 131072;
    o[sh * 128 + c]         = (xe + xo) * INV_SQRT2_;
    o[(sh + 512) * 128 + c] = (xe - xo) * INV_SQRT2_;
}

// ---------- 2. per-(b,c) mean / stdev (ddof=1) over s ----------
__global__ __launch_bounds__(128) void k_stats(const float* __restrict__ xw,
                                               float* __restrict__ meanv,
                                               float* __restrict__ stdv) {
    int b = blockIdx.x, c = threadIdx.x;
    const float* p = xw + (size_t)b * 131072 + c;
    float s = 0.f, ss = 0.f;
    for (int i = 0; i < 1024; ++i) { float v = p[i * 128]; s += v; ss += v * v; }
    float mn  = s * (1.0f / 1024.0f);
    float var = (ss - s * mn) * (1.0f / 1023.0f);
    meanv[b * 128 + c] = mn;
    stdv [b * 128 + c] = sqrtf(var + 1e-5f);
}

// ---------- 3. MLP: (B,S,4) -> LN+ReLU(32) -> LN+ReLU(16) ----------
__global__ __launch_bounds__(256) void k_mlp(const float* __restrict__ xm,
    const float* __restrict__ W1, const float* __restrict__ b1,
    const float* __restrict__ g1, const float* __restrict__ be1,
    const float* __restrict__ W2, const float* __restrict__ b2,
    const float* __restrict__ g2, const float* __restrict__ be2,
    float* __restrict__ h2out) {
    __shared__ float sW1[128], sb1[32], sg1[32], sbe1[32];
    __shared__ float sW2[512], sb2[16], sg2[16], sbe2[16];
    int tid = threadIdx.x;
    if (tid < 128) sW1[tid] = W1[tid];
    if (tid < 32) { sb1[tid] = b1[tid]; sg1[tid] = g1[tid]; sbe1[tid] = be1[tid]; }
    for (int i = tid; i < 512; i += 256) sW2[i] = W2[i];
    if (tid < 16) { sb2[tid] = b2[tid]; sg2[tid] = g2[tid]; sbe2[tid] = be2[tid]; }
    __syncthreads();
    size_t bs = (size_t)blockIdx.x * 256 + tid; // < 131072
    float x0 = xm[bs*4+0], x1 = xm[bs*4+1], x2 = xm[bs*4+2], x3 = xm[bs*4+3];
    float h1[32]; float s1 = 0.f;
    #pragma unroll
    for (int i = 0; i < 32; ++i) {
        float v = sb1[i] + sW1[i*4]*x0 + sW1[i*4+1]*x1 + sW1[i*4+2]*x2 + sW1[i*4+3]*x3;
        h1[i] = v; s1 += v;
    }
    float m1 = s1 * (1.f/32.f), vs = 0.f;
    #pragma unroll
    for (int i = 0; i < 32; ++i) { float d = h1[i] - m1; vs += d * d; }
    float inv = rsqrtf(vs * (1.f/32.f) + 1e-5f);
    #pragma unroll
    for (int i = 0; i < 32; ++i) {
        float v = (h1[i] - m1) * inv * sg1[i] + sbe1[i];
        h1[i] = v > 0.f ? v : 0.f;
    }
    float h2v[16]; float s2 = 0.f;
    #pragma unroll
    for (int j = 0; j < 16; ++j) {
        float v = sb2[j];
        #pragma unroll
        for (int i = 0; i < 32; ++i) v += sW2[j*32 + i] * h1[i];
        h2v[j] = v; s2 += v;
    }
    float m2 = s2 * (1.f/16.f), vs2 = 0.f;
    #pragma unroll
    for (int j = 0; j < 16; ++j) { float d = h2v[j] - m2; vs2 += d * d; }
    float inv2 = rsqrtf(vs2 * (1.f/16.f) + 1e-5f);
    #pragma unroll
    for (int j = 0; j < 16; ++j) {
        float v = (h2v[j] - m2) * inv2 * sg2[j] + sbe2[j];
        h2out[bs*16 + j] = v > 0.f ? v : 0.f;
    }
}

// ---------- 4. Wc(O,I,K) -> Wct[k][o][i] ----------
__global__ __launch_bounds__(256) void k_twc(const float* __restrict__ wc,
                                             float* __restrict__ wct) {
    size_t idx = (size_t)blockIdx.x * 256 + threadIdx.x; // 3*1024*1024
    size_t k  = idx >> 20;
    size_t oi = idx & 1048575;
    wct[idx] = wc[oi * 3 + k];
}

// ---------- 5. row sums: Rh[p]=sum Whp, Rt[p]=sum Wtp, Rbc[p]=sum Wtp*bc ----------
__global__ __launch_bounds__(256) void k_rsum(const float* __restrict__ whp,
                                              const float* __restrict__ wtp,
                                              const float* __restrict__ bc,
                                              float* __restrict__ rh,
                                              float* __restrict__ rt,
                                              float* __restrict__ rbc) {
    int p = blockIdx.x * 256 + threadIdx.x;
    if (p >= 336) return;
    const float* wh = whp + (size_t)p * 1024;
    const float* wt = wtp + (size_t)p * 1024;
    float a = 0.f, t = 0.f, c = 0.f;
    for (int s = 0; s < 1024; ++s) { a += wh[s]; float w = wt[s]; t += w; c += w * bc[s]; }
    rh[p] = a; rt[p] = t; rbc[p] = c;
}

// ---------- 6. conv as 3 shifted GEMMs with zero-padded LDS tile ----------
// cv[b,o,t] = sum_{i,k} Wct[k][o][i] * h2[b,i,t+k-1]
__global__ __launch_bounds__(256) void k_conv(const float* __restrict__ wct,
                                              const float* __restrict__ h2,
                                              float* __restrict__ cv) {
    // half of h2[b], 512 rows x 18 cols (cols 0 and 17 are the SAME-padding zeros)
    __shared__ float hp[512 * 18]; // 36 KB
    int b   = blockIdx.x >> 3;
    int grp = blockIdx.x & 7;
    int lane = threadIdx.x & 31;
    int w    = threadIdx.x >> 5;
    int o0   = (grp * 8 + w) * 16;
    int m    = lane & 15;
    int K0   = (lane >> 4) * 2;
    v8f acc = {0,0,0,0,0,0,0,0};
    for (int half = 0; half < 2; ++half) {
        __syncthreads(); // protect previous half's reads
        for (int idx = threadIdx.x; idx < 8192; idx += 256) {
            int i = idx >> 4, t = idx & 15;
            hp[i * 18 + 1 + t] =
                h2[(size_t)b * 16384 + (size_t)(half * 512 + i) * 16 + t];
        }
        for (int i = threadIdx.x; i < 512; i += 256) {
            hp[i * 18]      = 0.0f;
            hp[i * 18 + 17] = 0.0f;
        }
        __syncthreads();
        #pragma unroll
        for (int k = 0; k < 3; ++k) {
            const float* Ab = wct + (size_t)k * 1048576 + (size_t)(o0 + m) * 1024
                              + (size_t)half * 512 + K0;
            // hp[(i0+K0)*18 + 1 + (m + k - 1)]  -> unconditional ds_load
            const float* Bs = hp + (size_t)K0 * 18 + m + k;
            #pragma unroll 8
            for (int i0 = 0; i0 < 512; i0 += 4) {
                v2f a = *(const v2f*)(Ab + i0);
                v2f bv;
                bv.x = Bs[i0 * 18];
                bv.y = Bs[i0 * 18 + 18];
                acc = wmma4(a, bv, acc);
            }
        }
    }
    int mhi = (lane >> 4) * 8;
    #pragma unroll
    for (int r = 0; r < 8; ++r) {
        int o = o0 + r + mhi;
        cv[((size_t)b * 1024 + o) * 16 + m] = acc[r];
    }
}

// ---------- 7. Mt[b,p,g] = sum_s cv[b,s,g]*Wtp[p,s] + Rbc[p] ----------
__global__ __launch_bounds__(32) void k_mt(const float* __restrict__ wtp,
                                           const float* __restrict__ cv,
                                           const float* __restrict__ rbc,
                                           float* __restrict__ mt) {
    int b  = blockIdx.x / 21;
    int pt = blockIdx.x % 21;
    int p0 = pt * 16;
    int lane = threadIdx.x;
    int m  = lane & 15;
    int K0 = (lane >> 4) * 2;
    const float* Ab = wtp + (size_t)(p0 + m) * 1024 + K0;
    const float* Bb = cv + (size_t)b * 16384 + (size_t)K0 * 16 + m;
    v8f acc = {0,0,0,0,0,0,0,0};
    #pragma unroll 8
    for (int i0 = 0; i0 < 1024; i0 += 4) {
        v2f a = *(const v2f*)(Ab + i0);
        v2f bv;
        bv.x = Bb[i0 * 16];
        bv.y = Bb[i0 * 16 + 16];
        acc = wmma4(a, bv, acc);
    }
    int mhi = (lane >> 4) * 8;
    #pragma unroll
    for (int r = 0; r < 8; ++r) {
        int p = p0 + r + mhi;
        mt[((size_t)b * 336 + p) * 16 + m] = acc[r] + rbc[p];
    }
}

// ---------- 8. final: Gx GEMM + Mt@W3^T + RevIN/beta epilogue ----------
__global__ __launch_bounds__(32) void k_final(const float* __restrict__ whp,
    const float* __restrict__ xw, const float* __restrict__ mt,
    const float* __restrict__ w3, const float* __restrict__ meanv,
    const float* __restrict__ stdv, const float* __restrict__ rh,
    const float* __restrict__ rt, const float* __restrict__ b3,
    const float* __restrict__ btp, const float* __restrict__ bhp,
    float* __restrict__ out) {
    int blk = blockIdx.x;
    int b   = blk / 168;          // 21 p-tiles * 8 c-tiles
    int rem = blk % 168;
    int pt = rem / 8, ct = rem % 8;
    int p0 = pt * 16, c0 = ct * 16;
    int lane = threadIdx.x;
    int m  = lane & 15;
    int K0 = (lane >> 4) * 2;

    // Gx = xw @ Whp^T  (K = S = 1024)
    const float* Ab = whp + (size_t)(p0 + m) * 1024 + K0;
    const float* Bb = xw + (size_t)b * 131072 + (size_t)K0 * 128 + c0 + m;
    v8f accx = {0,0,0,0,0,0,0,0};
    #pragma unroll 8
    for (int i0 = 0; i0 < 1024; i0 += 4) {
        v2f a = *(const v2f*)(Ab + i0);
        v2f bv;
        bv.x = Bb[i0 * 128];
        bv.y = Bb[i0 * 128 + 128];
        accx = wmma4(a, bv, accx);
    }
    // Gt = Mt @ W3^T  (K = 16)
    const float* Am = mt + ((size_t)b * 336 + p0 + m) * 16 + K0;
    const float* Bw = w3 + (size_t)(c0 + m) * 16 + K0;
    v8f acct = {0,0,0,0,0,0,0,0};
    #pragma unroll
    for (int g0 = 0; g0 < 16; g0 += 4) {
        v2f a  = *(const v2f*)(Am + g0);
        v2f bv = *(const v2f*)(Bw + g0);
        acct = wmma4(a, bv, acct);
    }
    int c = c0 + m;
    float mn  = meanv[b * 128 + c];
    float sd  = stdv [b * 128 + c];
    float b3c = b3[c];
    int mhi = (lane >> 4) * 8;
    #pragma unroll
    for (int r = 0; r < 8; ++r) {
        int p = p0 + r + mhi;
        float t_out = acct[r] + b3c * rt[p] + btp[p];
        float val = BETA_ * (accx[r] - mn * rh[p])
                  + sd * (BETA_ * bhp[p] + (1.0f - BETA_) * t_out)
                  + mn;
        out[((size_t)b * 336 + p) * 128 + c] = val;
    }
}

extern "C" void kernel_launch(void* const* d_in, const int* in_sizes, int n_in,
                              void* d_out, int out_size, void* d_ws, size_t ws_size,
                              hipStream_t stream) {
    (void)in_sizes; (void)n_in; (void)out_size; (void)ws_size;
    const float* x   = (const float*)d_in[0];
    const float* xme = (const float*)d_in[1];
    const float* W1  = (const float*)d_in[4];
    const float* b1  = (const float*)d_in[5];
    const float* g1  = (const float*)d_in[6];
    const float* be1 = (const float*)d_in[7];
    const float* W2  = (const float*)d_in[8];
    const float* b2  = (const float*)d_in[9];
    const float* g2  = (const float*)d_in[10];
    const float* be2 = (const float*)d_in[11];
    const float* Wc  = (const float*)d_in[12];
    const float* bc  = (const float*)d_in[13];
    const float* W3  = (const float*)d_in[14];
    const float* b3  = (const float*)d_in[15];
    const float* Wtp = (const float*)d_in[16];
    const float* btp = (const float*)d_in[17];
    const float* Whp = (const float*)d_in[18];
    const float* bhp = (const float*)d_in[19];

    float* ws    = (float*)d_ws;
    float* xw    = ws + O_XW;
    float* wct   = ws + O_WCT;
    float* h2    = ws + O_H2;
    float* cv    = ws + O_CV;
    float* mtb   = ws + O_MT;
    float* meanv = ws + O_MEAN;
    float* stdv  = ws + O_STD;
    float* rh    = ws + O_RH;
    float* rt    = ws + O_RT;
    float* rbc   = ws + O_RBC;

    k_haar <<<32768, 256, 0, stream>>>(x, xw);
    k_stats<<<  128, 128, 0, stream>>>(xw, meanv, stdv);
    k_mlp  <<<  512, 256, 0, stream>>>(xme, W1, b1, g1, be1, W2, b2, g2, be2, h2);
    k_twc  <<<12288, 256, 0, stream>>>(Wc, wct);
    k_rsum <<<    2, 256, 0, stream>>>(Whp, Wtp, bc, rh, rt, rbc);
    k_conv <<< 1024, 256, 0, stream>>>(wct, h2, cv);
    k_mt   <<< 2688,  32, 0, stream>>>(Wtp, cv, rbc, mtb);
    k_final<<<21504,  32, 0, stream>>>(Whp, xw, mtb, W3, meanv, stdv,
                                       rh, rt, b3, btp, bhp, (float*)d_out);
}